// MultiHeadDiffAttention_17849884082433
// MI455X (gfx1250) — compile-verified
//
#include <hip/hip_runtime.h>
#include <hip/hip_bf16.h>
#include <math.h>

// ---------------- problem constants ----------------
#define Bc 8
#define Nc 1024
#define Ec 256
#define Hc 4
#define Uc 6
#define HD2 64                      // 2*head_dim
#define LAMBDA_INIT_F 0.6321205588285577f   // 1 - exp(-1)

typedef __attribute__((ext_vector_type(2))) float v2f;
typedef __attribute__((ext_vector_type(8))) float v8f;

// pointee type the async-LDS builtins expect (clang 'V4i'): vector_size(16) int
typedef int v4i_async __attribute__((vector_size(16)));

// ---- CDNA5 async global->LDS path (gfx1250), guarded so fallback still compiles
#if defined(__has_builtin)
#if __has_builtin(__builtin_amdgcn_global_load_async_to_lds_b128) && \
    __has_builtin(__builtin_amdgcn_s_wait_asynccnt)
#define USE_ASYNC_LDS 1
#endif
#endif
#ifndef USE_ASYNC_LDS
#define USE_ASYNC_LDS 0
#endif

// =====================================================================
// beta = sigmoid( exp(sum(lq*lk)) * LAMBDA_INIT )
// =====================================================================
__global__ __launch_bounds__(128) void beta_kernel(const float* __restrict__ lq,
                                                   const float* __restrict__ lk,
                                                   float* __restrict__ beta_ws,
                                                   float* __restrict__ beta_out) {
    __shared__ float red[128];
    float s = 0.f;
    for (int i = threadIdx.x; i < 100; i += 128) s += lq[i] * lk[i];
    red[threadIdx.x] = s;
    __syncthreads();
    for (int off = 64; off > 0; off >>= 1) {
        if (threadIdx.x < off) red[threadIdx.x] += red[threadIdx.x + off];
        __syncthreads();
    }
    if (threadIdx.x == 0) {
        float l1 = __expf(red[0]);
        float b  = 1.f / (1.f + __expf(-l1 * LAMBDA_INIT_F));
        *beta_ws  = b;
        *beta_out = b;
    }
}

// =====================================================================
// Generic fp32 WMMA GEMM:  Y = X[M,256] * W[F=256,K=256]^T
// MODE 0: Y[m*256 + f]                     (row-major [M,256])
// MODE 1: Y[((b*H + h)*N + n)*64 + d]      (v layout, f = h*64+d, m = b*N+n)
// Block: 128 threads = 4 waves. Block tile: 64 rows x 16 cols.
// W tile (16 f x 256 e) staged transposed in LDS: wt[e*16 + f].
// =====================================================================
template <int MODE>
__global__ __launch_bounds__(128) void gemm_proj_kernel(const float* __restrict__ X,
                                                        const float* __restrict__ W,
                                                        float* __restrict__ Y) {
    __shared__ float wt[256 * 16];      // 16 KB
    const int t  = threadIdx.x;
    const int m0 = blockIdx.x * 64;
    const int f0 = blockIdx.y * 16;

    // --- stage W tile: 16 rows of W (each 256 floats, coalesced), transpose into LDS
    {
        const int fr = t >> 3;          // 0..15 : which f row
        const int r8 = t & 7;           // 0..7  : position within row
        const float* wrow = W + (size_t)(f0 + fr) * 256 + r8 * 32;
        float* dcol = wt + fr;
#pragma unroll
        for (int q = 0; q < 8; ++q) {
            float4 w4 = *(const float4*)(wrow + q * 4);
            const int e = r8 * 32 + q * 4;
            dcol[(e + 0) * 16] = w4.x;
            dcol[(e + 1) * 16] = w4.y;
            dcol[(e + 2) * 16] = w4.z;
            dcol[(e + 3) * 16] = w4.w;
        }
    }
    __syncthreads();

    const int wv   = t >> 5;            // wave 0..3 -> 16-row sub-tile
    const int lane = t & 31;
    const int row  = lane & 15;         // A row / B-C-D column
    const int kh   = lane >> 4;         // K half (0 -> K{0,1}, 1 -> K{2,3})

    const float* xp = X + (size_t)(m0 + wv * 16 + row) * 256;
    v8f acc = {};
#pragma unroll 8
    for (int k = 0; k < 256; k += 4) {
        const int kk = k + kh * 2;
        v2f a = *(const v2f*)(xp + kk);                 // A[row, kk..kk+1]
        v2f b;
        b[0] = wt[(kk + 0) * 16 + row];                 // B[kk,   col]
        b[1] = wt[(kk + 1) * 16 + row];                 // B[kk+1, col]
        acc = __builtin_amdgcn_wmma_f32_16x16x4_f32(false, a, false, b,
                                                    (short)0, acc, false, false);
    }

    // --- store D: element acc[vI] is (M = vI + 8*kh, N = col) of the 16x16 tile
    const int col = row;
#pragma unroll
    for (int vI = 0; vI < 8; ++vI) {
        const int mrow = m0 + wv * 16 + vI + 8 * kh;
        const int f    = f0 + col;
        const float val = acc[vI];
        if (MODE == 0) {
            Y[(size_t)mrow * 256 + f] = val;
        } else {
            const int b = mrow >> 10, n = mrow & 1023;
            const int h = f >> 6,     d = f & 63;
            Y[(((size_t)b * Hc + h) * Nc + n) * HD2 + d] = val;
        }
    }
}

// =====================================================================
// Fused: scores = u * u_w^T + u_b ; mask ; softmax(j) ; diff -> attn_w
// One 256-thread block per (b, i) row; each thread owns 4 j's.
// u:     [B, 6, N, N]  (read once, coalesced)
// umask: [B, N, N] bool bytes, mask[b,0,i,j] = umask[b, j, i]
// attn_w out: [B, H, N, N]
// =====================================================================
__global__ __launch_bounds__(256) void scores_softmax_kernel(
        const float* __restrict__ u, const unsigned char* __restrict__ umask,
        const float* __restrict__ u_w, const float* __restrict__ u_b,
        const float* __restrict__ beta_p, float* __restrict__ attn_out) {
    __shared__ float w[48];
    __shared__ float bia[8];
    __shared__ float red[8][256];       // 8 KB reduction scratch

    const int blk = blockIdx.x;
    const int b   = blk >> 10;
    const int i   = blk & 1023;
    const int t   = threadIdx.x;
    const int j0  = t * 4;

    if (t < 48) w[t] = u_w[t];
    if (t < 8)  bia[t] = u_b[t];
    __syncthreads();

    // --- channel mix: s[o][jj] = sum_c u_w[o,c] * u[b,c,i,j] + u_b[o]
    float s[8][4];
#pragma unroll
    for (int o = 0; o < 8; ++o)
#pragma unroll
        for (int jj = 0; jj < 4; ++jj) s[o][jj] = bia[o];

#pragma unroll
    for (int c = 0; c < Uc; ++c) {
        const float* up = u + (((size_t)b * Uc + c) * Nc + i) * Nc + j0;
        float4 uv = *(const float4*)up;
        float uvv[4] = {uv.x, uv.y, uv.z, uv.w};
#pragma unroll
        for (int o = 0; o < 8; ++o) {
            const float wc = w[o * 6 + c];
#pragma unroll
            for (int jj = 0; jj < 4; ++jj) s[o][jj] = fmaf(wc, uvv[jj], s[o][jj]);
        }
    }

    // --- mask: umask[b, j, i]
#pragma unroll
    for (int jj = 0; jj < 4; ++jj) {
        const unsigned char m = umask[((size_t)b * Nc + (j0 + jj)) * Nc + i];
        if (m) {
#pragma unroll
            for (int o = 0; o < 8; ++o) s[o][jj] = -__builtin_inff();
        }
    }

    // --- row max (8 channels at once)
    float cmax[8];
#pragma unroll
    for (int o = 0; o < 8; ++o) {
        float m = s[o][0];
#pragma unroll
        for (int jj = 1; jj < 4; ++jj) m = fmaxf(m, s[o][jj]);
        red[o][t] = m;
    }
    __syncthreads();
    for (int off = 128; off > 0; off >>= 1) {
        if (t < off)
#pragma unroll
            for (int o = 0; o < 8; ++o)
                red[o][t] = fmaxf(red[o][t], red[o][t + off]);
        __syncthreads();
    }
#pragma unroll
    for (int o = 0; o < 8; ++o) cmax[o] = red[o][0];
    __syncthreads();

    // --- exp + row sum
    float csum[8];
#pragma unroll
    for (int o = 0; o < 8; ++o) {
        float lsum = 0.f;
#pragma unroll
        for (int jj = 0; jj < 4; ++jj) {
            s[o][jj] = __expf(s[o][jj] - cmax[o]);   // exp(-inf) = 0 on masked
            lsum += s[o][jj];
        }
        red[o][t] = lsum;
    }
    __syncthreads();
    for (int off = 128; off > 0; off >>= 1) {
        if (t < off)
#pragma unroll
            for (int o = 0; o < 8; ++o)
                red[o][t] += red[o][t + off];
        __syncthreads();
    }
#pragma unroll
    for (int o = 0; o < 8; ++o) csum[o] = red[o][0];

    // --- differential combine + store
    const float beta = *beta_p;
#pragma unroll
    for (int h = 0; h < Hc; ++h) {
        const float inv0 = 1.f / csum[2 * h];
        const float inv1 = 1.f / csum[2 * h + 1];
        float4 outv;
        outv.x = s[2 * h][0] * inv0 - beta * s[2 * h + 1][0] * inv1;
        outv.y = s[2 * h][1] * inv0 - beta * s[2 * h + 1][1] * inv1;
        outv.z = s[2 * h][2] * inv0 - beta * s[2 * h + 1][2] * inv1;
        outv.w = s[2 * h][3] * inv0 - beta * s[2 * h + 1][3] * inv1;
        *(float4*)(attn_out + (((size_t)b * Hc + h) * Nc + i) * Nc + j0) = outv;
    }
}

// =====================================================================
// ctx[b, n, h*64+d] = sum_j attn_w[b,h,n,j] * v[b,h,j,d]
// Block: 128 threads = 4 waves; tile = 16 rows (i) x 64 cols (d).
// A tile (16 x 1024) staged in LDS in 2 chunks of 512 (pitch 516 -> no
// bank conflicts: rows step 4 banks, lane halves step 2).
// Staging uses gfx1250 async global->LDS b128 copies (ASYNCcnt) when the
// builtin is available: no VGPR round-trip, copy overlaps compute tail.
// =====================================================================
__global__ __launch_bounds__(128) void attn_ctx_kernel(const float* __restrict__ attn,
                                                       const float* __restrict__ v,
                                                       float* __restrict__ ctx) {
    __shared__ float at[16 * 516];      // ~33 KB

    const int i0 = blockIdx.x * 16;
    const int h  = blockIdx.y;
    const int b  = blockIdx.z;
    const int t  = threadIdx.x;

    const float* ap = attn + (((size_t)b * Hc + h) * Nc + i0) * Nc;
    const float* vbase = v + ((size_t)b * Hc + h) * Nc * HD2;

    const int wv   = t >> 5;
    const int lane = t & 31;
    const int row  = lane & 15;
    const int kh   = lane >> 4;
    const int col  = row;
    const float* vp = vbase + wv * 16 + col;

    const int rr = t >> 3;              // 0..15 : row of the A tile
    const int c8 = t & 7;               // 0..7  : 64-float segment within row

    v8f acc = {};
    for (int kc = 0; kc < Nc; kc += 512) {
        // stage 16 x 512 chunk of attn_w
        const float* arow = ap + (size_t)rr * Nc + kc + c8 * 64;
        float* dst = at + rr * 516 + c8 * 64;
#if USE_ASYNC_LDS
        {
            v4i_async* gsrc = (v4i_async*)arow;   // generic ptr, right pointee type;
            v4i_async* ldst = (v4i_async*)dst;    // clang converts to __device__/__shared__
#pragma unroll
            for (int q = 0; q < 4; ++q)
                __builtin_amdgcn_global_load_async_to_lds_b128(gsrc + q, ldst + q, 0, 0);
        }
        __builtin_amdgcn_s_wait_asynccnt(0);
#else
#pragma unroll
        for (int q = 0; q < 16; ++q)
            *(float4*)(dst + q * 4) = *(const float4*)(arow + q * 4);
#endif
        __syncthreads();

#pragma unroll 4
        for (int k = 0; k < 512; k += 4) {
            const int kk = k + kh * 2;
            v2f a = *(const v2f*)(at + row * 516 + kk);          // A[row, kk..kk+1]
            v2f bvec;
            bvec[0] = vp[(size_t)(kc + kk + 0) * HD2];           // B[kk,   col]
            bvec[1] = vp[(size_t)(kc + kk + 1) * HD2];           // B[kk+1, col]
            acc = __builtin_amdgcn_wmma_f32_16x16x4_f32(false, a, false, bvec,
                                                        (short)0, acc, false, false);
        }
        __syncthreads();
    }

#pragma unroll
    for (int vI = 0; vI < 8; ++vI) {
        const int n = i0 + vI + 8 * kh;
        ctx[((size_t)b * Nc + n) * Ec + h * HD2 + wv * 16 + col] = acc[vI];
    }
}

// =====================================================================
// host-side launcher
// =====================================================================
extern "C" void kernel_launch(void* const* d_in, const int* in_sizes, int n_in,
                              void* d_out, int out_size, void* d_ws, size_t ws_size,
                              hipStream_t stream) {
    const float*         x     = (const float*)d_in[0];
    const float*         u     = (const float*)d_in[1];
    const unsigned char* umask = (const unsigned char*)d_in[2];   // jax bool -> 1 byte
    const float*         v_w   = (const float*)d_in[3];
    const float*         out_w = (const float*)d_in[4];
    const float*         u_w   = (const float*)d_in[5];
    const float*         u_b   = (const float*)d_in[6];
    const float*         lq    = (const float*)d_in[7];
    const float*         lk    = (const float*)d_in[8];

    float* out      = (float*)d_out;
    float* attn_out = out + (size_t)Bc * Nc * Ec;                 // 2,097,152
    float* beta_out = attn_out + (size_t)Bc * Hc * Nc * Nc;       // +33,554,432

    char*  wsb     = (char*)d_ws;
    float* beta_ws = (float*)wsb;
    float* v_ws    = (float*)(wsb + 256);                                  // 8 MB
    float* ctx_ws  = (float*)(wsb + 256 + (size_t)Bc * Hc * Nc * HD2 * 4); // 8 MB

    beta_kernel<<<1, 128, 0, stream>>>(lq, lk, beta_ws, beta_out);

    // v = x * v_w^T  -> [B,H,N,64]
    gemm_proj_kernel<1><<<dim3((Bc * Nc) / 64, Ec / 16), 128, 0, stream>>>(x, v_w, v_ws);

    // fused conv1x1 + mask + softmax + differential -> attn_w (also an output)
    scores_softmax_kernel<<<Bc * Nc, 256, 0, stream>>>(u, umask, u_w, u_b, beta_ws, attn_out);

    // ctx = attn_w @ v  -> [B,N,256]
    attn_ctx_kernel<<<dim3(Nc / 16, Hc, Bc), 128, 0, stream>>>(attn_out, v_ws, ctx_ws);

    // out = ctx * out_w^T
    gemm_proj_kernel<0><<<dim3((Bc * Nc) / 64, Ec / 16), 128, 0, stream>>>(ctx_ws, out_w, out);
}